// TensorProductConvolution_84267258348035
// MI455X (gfx1250) — compile-verified
//
#include <hip/hip_runtime.h>

// Problem constants (fixed by the reference)
#define N_NODES 50000
#define N_EDGES 800000
#define FEAT    128      // 4*MUL
#define MSG     256      // 8*MUL message/agg width
#define NTILES  (N_NODES / 16)   // 3125, exact

static constexpr float SQRT3 = 1.7320508075688772f;
static constexpr float INV_NORM = 0.125f; // 1/sqrt(2*MUL) = 1/sqrt(64)

typedef float v2f __attribute__((ext_vector_type(2)));
typedef float v8f __attribute__((ext_vector_type(8)));

// ---------------------------------------------------------------------------
// Kernel 1: zero the edge-aggregation workspace (atomics accumulate into it)
// ---------------------------------------------------------------------------
__global__ void zero_agg_kernel(float4* __restrict__ agg4, int n4) {
    int i = blockIdx.x * blockDim.x + threadIdx.x;
    if (i < n4) agg4[i] = make_float4(0.f, 0.f, 0.f, 0.f);
}

// ---------------------------------------------------------------------------
// Kernel 2: edge message + scatter-add.  One wave32 per edge; lane == channel.
// Edge id is forced into an SGPR so the index / position reads become scalar
// (SMEM) loads and the atomics use the SADDR form.
//   msg layout per edge/node: [mid0(32) | mid1(96) | mid2(96) | mid3(32)]
// ---------------------------------------------------------------------------
__global__ void edge_kernel(const float* __restrict__ feats,
                            const float* __restrict__ pos,
                            const int*   __restrict__ esrc,
                            const int*   __restrict__ edst,
                            const float* __restrict__ tpw,
                            float*       __restrict__ agg) {
    const int lane = threadIdx.x & 31;
    const int e    = __builtin_amdgcn_readfirstlane(
                        (int)blockIdx.x * 8 + (int)(threadIdx.x >> 5));
    if (e >= N_EDGES) return;               // whole wave retires together

    const int si = esrc[e];                 // scalar loads (uniform address)
    const int di = edst[e];

    const float dx = pos[di * 3 + 0] - pos[si * 3 + 0];
    const float dy = pos[di * 3 + 1] - pos[si * 3 + 1];
    const float dz = pos[di * 3 + 2] - pos[si * 3 + 2];
    const float rinv = rsqrtf(dx * dx + dy * dy + dz * dz + 1e-12f);
    const float ux = dx * rinv, uy = dy * rinv, uz = dz * rinv;
    const float shx = SQRT3 * ux, shy = SQRT3 * uy, shz = SQRT3 * uz;

    const float* __restrict__ f = feats + (size_t)si * FEAT;
    const float s  = f[lane];               // scalar channel u == lane
    const float v0 = f[32 + lane * 3 + 0];  // vector channel u, comps 0..2
    const float v1 = f[32 + lane * 3 + 1];
    const float v2 = f[32 + lane * 3 + 2];

    const float w0 = tpw[lane];
    const float w1 = tpw[32 + lane];
    const float w2 = tpw[64 + lane];
    const float w3 = tpw[96 + lane];

    float* __restrict__ a = agg + (size_t)di * MSG;

    // mid0 = w0*s
    unsafeAtomicAdd(a + lane, w0 * s);
    // mid1 = (w1*s) outer sh1
    const float w1s = w1 * s;
    unsafeAtomicAdd(a + 32 + lane * 3 + 0, w1s * shx);
    unsafeAtomicAdd(a + 32 + lane * 3 + 1, w1s * shy);
    unsafeAtomicAdd(a + 32 + lane * 3 + 2, w1s * shz);
    // mid2 = w2 * v
    unsafeAtomicAdd(a + 128 + lane * 3 + 0, w2 * v0);
    unsafeAtomicAdd(a + 128 + lane * 3 + 1, w2 * v1);
    unsafeAtomicAdd(a + 128 + lane * 3 + 2, w2 * v2);
    // mid3 = w3 * (v . sh1)/sqrt3 == w3 * (v . unit)
    unsafeAtomicAdd(a + 224 + lane, w3 * (v0 * ux + v1 * uy + v2 * uz));
}

// ---------------------------------------------------------------------------
// Kernel 3: node-stage GEMMs via V_WMMA_F32_16X16X4_F32.
// blockIdx.y = gemm variant g (true SGPR -> no per-lane divergence):
//   g=0: [a0|a3] @ W_s.  A col = u        (u<32)  or  192+u      (u>=32)
//        -> stride-1 walk with a scalar +192 jump at kk==8 (u==32).
//   g=1..3: vector comp m=g-1, [a1|a2](:,:,m) @ W_v.
//        A col = 32 + 3u + m for ALL u (the u>=32 half collapses to the same
//        affine form: 128 + 3(u-32) + m == 32 + 3u + m) -> pure stride-3 walk.
// One wave per 16-node tile; two 16x16 accumulators cover the 32 output cols;
// K=64 walked in 16 WMMA steps of K=4.
//
// Fragment layouts (ISA 7.12.2, wave32):
//   A(16x4): lane%16 -> M, K = kbase + 2*(lane/16) + reg
//   B(4x16): lane%16 -> N, K = kbase + 2*(lane/16) + reg
//   D(16x16): lane%16 -> N, M = reg + 8*(lane/16)
// ---------------------------------------------------------------------------
__global__ void node_gemm_kernel(const float* __restrict__ agg,
                                 const float* __restrict__ Ws,
                                 const float* __restrict__ Wv,
                                 float*       __restrict__ out) {
    const int g    = (int)blockIdx.y;                 // scalar gemm variant
    const int tile = (int)blockIdx.x * 8 + (int)(threadIdx.x >> 5);
    if (tile >= NTILES) return;                       // whole wave retires

    const int lane = threadIdx.x & 31;
    const int half = lane >> 4;
    const int lm   = lane & 15;

    const float* __restrict__ B = (g == 0) ? Ws : Wv;
    const float* __restrict__ arow = agg + (size_t)(tile * 16 + lm) * MSG;

    // Per-lane A walk: base + stride*u, u advances by 4 per kk step.
    int astride, abase;
    if (g == 0) { astride = 1; abase = half * 2; }
    else        { astride = 3; abase = 32 + (g - 1) + 3 * (half * 2); }

    const float* __restrict__ aptr = arow + abase;
    const float* __restrict__ bptr = B + (half * 2) * 32 + lm;

    v8f acc0 = {0.f, 0.f, 0.f, 0.f, 0.f, 0.f, 0.f, 0.f};
    v8f acc1 = {0.f, 0.f, 0.f, 0.f, 0.f, 0.f, 0.f, 0.f};

#pragma unroll
    for (int kk = 0; kk < 16; ++kk) {
        if (kk == 8 && g == 0) aptr += 192;           // scalar jump a0 -> a3

        v2f a, b0, b1;
        a[0]  = aptr[0];
        a[1]  = aptr[astride];
        b0[0] = bptr[0];        // reg0: K = kbase+2*half,   N = lm
        b0[1] = bptr[32];       // reg1: K = kbase+2*half+1, N = lm
        b1[0] = bptr[16];       //                            N = 16+lm
        b1[1] = bptr[48];

        acc0 = __builtin_amdgcn_wmma_f32_16x16x4_f32(false, a, false, b0,
                                                     (short)0, acc0, false, false);
        acc1 = __builtin_amdgcn_wmma_f32_16x16x4_f32(false, a, false, b1,
                                                     (short)0, acc1, false, false);

        aptr += astride * 4;    // u += 4
        bptr += 128;            // 4 rows of 32
    }

    const int m = g - 1;
#pragma unroll
    for (int r = 0; r < 8; ++r) {
        const int M = r + half * 8;
        float* __restrict__ orow = out + (size_t)(tile * 16 + M) * FEAT;
        const float v0 = acc0[r] * INV_NORM;          // N = lm
        const float v1 = acc1[r] * INV_NORM;          // N = 16 + lm
        if (g == 0) {
            orow[lm]      = v0;
            orow[16 + lm] = v1;
        } else {
            orow[32 + lm * 3 + m]        = v0;
            orow[32 + (16 + lm) * 3 + m] = v1;
        }
    }
}

// ---------------------------------------------------------------------------
extern "C" void kernel_launch(void* const* d_in, const int* in_sizes, int n_in,
                              void* d_out, int out_size, void* d_ws, size_t ws_size,
                              hipStream_t stream) {
    const float* node_features = (const float*)d_in[0];
    const float* positions     = (const float*)d_in[1];
    const int*   edge_src      = (const int*)d_in[2];
    const int*   edge_dst      = (const int*)d_in[3];
    const float* tp_weight     = (const float*)d_in[4];
    const float* W_s           = (const float*)d_in[5];
    const float* W_v           = (const float*)d_in[6];
    float* out = (float*)d_out;

    // agg: N x 256 f32 = 51.2 MB in workspace (fits in the 192 MB L2 -> fast atomics)
    float* agg = (float*)d_ws;

    // 1) zero accumulator (required every call; atomics below accumulate)
    const int n4 = N_NODES * MSG / 4;                 // float4 count
    zero_agg_kernel<<<(n4 + 255) / 256, 256, 0, stream>>>((float4*)agg, n4);

    // 2) edge stage: one wave32 per edge, 8 edges per 256-thread block
    edge_kernel<<<(N_EDGES + 7) / 8, 256, 0, stream>>>(
        node_features, positions, edge_src, edge_dst, tp_weight, agg);

    // 3) node stage: WMMA GEMMs; grid.y selects the gemm variant (SGPR)
    dim3 ggrid((NTILES + 7) / 8, 4);
    node_gemm_kernel<<<ggrid, 256, 0, stream>>>(agg, W_s, W_v, out);
}